// Model_77610059038802
// MI455X (gfx1250) — compile-verified
//
#include <hip/hip_runtime.h>
#include <math.h>

#define N_NODES 100000
#define N_EDGES 1000000
#define DIM 64
#define NTILES (N_NODES / 16)   // 6250, exact

typedef __attribute__((ext_vector_type(2))) float v2f;
typedef __attribute__((ext_vector_type(8))) float v8f;

// d_out layout (flat float offsets, reference return order h1,h2,c1,c2,h3,h4)
#define H1_OFF 0
#define H2_OFF (N_NODES * DIM)
#define C1_OFF (2 * N_NODES * DIM)
#define C2_OFF (2 * N_NODES * DIM + DIM)
#define H3_OFF (2 * N_NODES * DIM + 2 * DIM)
#define H4_OFF (3 * N_NODES * DIM + 2 * DIM)
#define OUT_TOTAL (4 * N_NODES * DIM + 2 * DIM)

// ---------------------------------------------------------------------------
// Zero the whole output buffer (used as in-place aggregation scratch) and the
// 128-float column-sum accumulator in the workspace. float4 grid-stride.
// ---------------------------------------------------------------------------
__global__ void zero_kernel(float* __restrict__ out, float* __restrict__ acc) {
    const size_t n4 = OUT_TOTAL / 4;  // divisible: 25,600,128 / 4
    size_t i = (size_t)blockIdx.x * blockDim.x + threadIdx.x;
    float4 z = {0.f, 0.f, 0.f, 0.f};
    for (size_t j = i; j < n4; j += (size_t)gridDim.x * blockDim.x)
        ((float4*)out)[j] = z;
    if (i < 128) acc[i] = 0.f;
}

// ---------------------------------------------------------------------------
// Edge pass: one wave32 per edge. Lane l owns columns 2l, 2l+1 (one float2
// load per feature matrix = coalesced 256B per wave). Scatter-add into the
// two aggregation targets (feat -> aggF, shuf -> aggS) with f32 atomics that
// stay resident in the 192MB L2 (accumulators total 102MB).
// ---------------------------------------------------------------------------
__global__ void edge_kernel(const float* __restrict__ feat,
                            const float* __restrict__ shuf,
                            const int* __restrict__ src,
                            const int* __restrict__ dst,
                            float* __restrict__ aggF,
                            float* __restrict__ aggS) {
    const int lane   = threadIdx.x & 31;
    const int wave   = blockIdx.x * (blockDim.x >> 5) + (threadIdx.x >> 5);
    const int nwaves = gridDim.x * (blockDim.x >> 5);
    const int col    = lane * 2;

    for (int e = wave; e < N_EDGES; e += nwaves) {
        const int s = src[e];
        const int d = dst[e];
        const float2 f = *(const float2*)(feat + (size_t)s * DIM + col);
        const float2 g = *(const float2*)(shuf + (size_t)s * DIM + col);
        float* pF = aggF + (size_t)d * DIM + col;
        float* pS = aggS + (size_t)d * DIM + col;
        atomicAdd(pF + 0, f.x);
        atomicAdd(pF + 1, f.y);
        atomicAdd(pS + 0, g.x);
        atomicAdd(pS + 1, g.y);
    }
}

// ---------------------------------------------------------------------------
// In-place per-node linear layer with fp32 WMMA: h[n,:] = agg[n,:] @ W + b.
// One wave -> one 16-row tile x full 64 columns (4 accumulators of 16x16).
// A-frag (V_WMMA_F32_16X16X4_F32 layout): lanes 0-15 rows M=0..15 holding
// K={k0,k0+1}; lanes 16-31 same rows holding K={k0+2,k0+3}. B is W^T staged
// in LDS so each B-frag is one float2 LDS read. C layout: VGPR r = row
// (lane<16 ? r : 8+r), col = lane&15 within the 16-col tile.
// Optionally accumulates column sums (for the mean->sigmoid heads).
// ---------------------------------------------------------------------------
__global__ void gemm_kernel(float* __restrict__ h,          // agg in, h out (in place)
                            const float* __restrict__ W,    // [64 x 64], row = k
                            const float* __restrict__ bias, // [64]
                            float* __restrict__ csum) {     // [64] or nullptr
    __shared__ float Wt[DIM * DIM];  // transposed: Wt[n*64 + k] = W[k*64 + n]
    __shared__ float blds[DIM];

    for (int idx = threadIdx.x; idx < DIM * DIM; idx += blockDim.x) {
        const int k = idx >> 6, n = idx & 63;
        Wt[n * DIM + k] = W[idx];
    }
    if (threadIdx.x < DIM) blds[threadIdx.x] = bias[threadIdx.x];
    __syncthreads();

    const int lane = threadIdx.x & 31;
    const int wave = threadIdx.x >> 5;
    const int tile = blockIdx.x * 8 + wave;
    if (tile >= NTILES) return;  // uniform per wave -> EXEC all-1 inside

    const int m15  = lane & 15;
    const int kofs = (lane >> 4) * 2;               // 0 or 2
    const float* arow = h + (size_t)(tile * 16 + m15) * DIM;

    v8f c0 = {}, c1 = {}, c2 = {}, c3 = {};
#pragma unroll
    for (int k0 = 0; k0 < DIM; k0 += 4) {
        const v2f a  = *(const v2f*)(arow + k0 + kofs);
        const v2f b0 = *(const v2f*)(&Wt[(m15 + 0) * DIM + k0 + kofs]);
        const v2f b1 = *(const v2f*)(&Wt[(m15 + 16) * DIM + k0 + kofs]);
        const v2f b2 = *(const v2f*)(&Wt[(m15 + 32) * DIM + k0 + kofs]);
        const v2f b3 = *(const v2f*)(&Wt[(m15 + 48) * DIM + k0 + kofs]);
        c0 = __builtin_amdgcn_wmma_f32_16x16x4_f32(false, a, false, b0, (short)0, c0, false, false);
        c1 = __builtin_amdgcn_wmma_f32_16x16x4_f32(false, a, false, b1, (short)0, c1, false, false);
        c2 = __builtin_amdgcn_wmma_f32_16x16x4_f32(false, a, false, b2, (short)0, c2, false, false);
        c3 = __builtin_amdgcn_wmma_f32_16x16x4_f32(false, a, false, b3, (short)0, c3, false, false);
    }

    const int rbase = tile * 16 + (lane >> 4) * 8;  // this lane-half's first row
#pragma unroll
    for (int nt = 0; nt < 4; ++nt) {
        const int n = nt * 16 + m15;
        const float bn = blds[n];
        const v8f c = (nt == 0) ? c0 : (nt == 1) ? c1 : (nt == 2) ? c2 : c3;
        float colsum = 0.f;
#pragma unroll
        for (int r = 0; r < 8; ++r) {
            const float v = c[r] + bn;
            h[(size_t)(rbase + r) * DIM + n] = v;
            colsum += v;
        }
        if (csum) atomicAdd(&csum[n], colsum);  // partial column sum (8 rows)
    }
}

// ---------------------------------------------------------------------------
// c = sigmoid(colsum / N)
// ---------------------------------------------------------------------------
__global__ void finish_kernel(const float* __restrict__ acc,
                              float* __restrict__ c1, float* __restrict__ c2) {
    const int j = threadIdx.x;
    if (j < 64) {
        const float m = acc[j] * (1.0f / (float)N_NODES);
        c1[j] = 1.0f / (1.0f + __expf(-m));
    } else if (j < 128) {
        const float m = acc[j] * (1.0f / (float)N_NODES);
        c2[j - 64] = 1.0f / (1.0f + __expf(-m));
    }
}

extern "C" void kernel_launch(void* const* d_in, const int* in_sizes, int n_in,
                              void* d_out, int out_size, void* d_ws, size_t ws_size,
                              hipStream_t stream) {
    const float* feat = (const float*)d_in[0];
    const float* shuf = (const float*)d_in[1];
    const int*   src1 = (const int*)d_in[2];
    const int*   dst1 = (const int*)d_in[3];
    const int*   src2 = (const int*)d_in[4];
    const int*   dst2 = (const int*)d_in[5];
    const float* W1   = (const float*)d_in[6];
    const float* b1   = (const float*)d_in[7];
    const float* W2   = (const float*)d_in[8];
    const float* b2   = (const float*)d_in[9];

    float* out = (float*)d_out;
    float* acc = (float*)d_ws;  // 128 floats: [0:64]=h1 colsums, [64:128]=h2

    float* h1 = out + H1_OFF;
    float* h2 = out + H2_OFF;
    float* h3 = out + H3_OFF;
    float* h4 = out + H4_OFF;

    // 1) zero aggregation targets + colsum accumulators
    zero_kernel<<<4096, 256, 0, stream>>>(out, acc);

    // 2) edge passes: graph1 feeds h1(feat)+h3(shuf), graph2 feeds h2+h4
    edge_kernel<<<2048, 256, 0, stream>>>(feat, shuf, src1, dst1, h1, h3);
    edge_kernel<<<2048, 256, 0, stream>>>(feat, shuf, src2, dst2, h2, h4);

    // 3) in-place WMMA linear layers (h1/h2 also produce column sums)
    const int gblocks = (NTILES + 7) / 8;  // 8 waves (tiles) per 256-thread block
    gemm_kernel<<<gblocks, 256, 0, stream>>>(h1, W1, b1, acc);
    gemm_kernel<<<gblocks, 256, 0, stream>>>(h2, W2, b2, acc + 64);
    gemm_kernel<<<gblocks, 256, 0, stream>>>(h3, W1, b1, nullptr);
    gemm_kernel<<<gblocks, 256, 0, stream>>>(h4, W2, b2, nullptr);

    // 4) sigmoid(mean) heads
    finish_kernel<<<1, 128, 0, stream>>>(acc, out + C1_OFF, out + C2_OFF);
}